// RSConv_55448027791744
// MI455X (gfx1250) — compile-verified
//
#include <hip/hip_runtime.h>
#include <hip/hip_bf16.h>

typedef __attribute__((ext_vector_type(2))) float v2f;
typedef __attribute__((ext_vector_type(8))) float v8f;

#define BATCH 4
#define CIN   256
#define HH    128
#define WW    128
#define HW    (HH * WW)          // 16384
#define NPIX  (BATCH * HW)       // 65536
#define OC    32
#define TILEW 64
#define NBLK_CONV (BATCH * HH * (WW / TILEW))   // 1024
#define XSTRIDE 67               // LDS col stride (3c mod 64 conflict-free)

// ---------------------------------------------------------------------------
// K0: orientation-bin gather  feats[NCHW] -> xbuf[NHWC(32)] + idx per pixel
// ---------------------------------------------------------------------------
__global__ __launch_bounds__(256) void gather_k(const float* __restrict__ feats,
                                                const float* __restrict__ rois,
                                                float* __restrict__ xbuf,
                                                int* __restrict__ idxb) {
  int p  = blockIdx.x * 256 + threadIdx.x;     // 0..65535
  int b  = p >> 14;                            // / 16384
  int hw = p & 16383;
  float ang = rois[(size_t)p * 5 + 4];
  int a = (int)floorf((ang + 0.7853981633974483f) * (8.0f / 3.14159265358979323846f));
  a = a < 0 ? 0 : (a > 7 ? 7 : a);
  idxb[p] = a;
  const float* src = feats + ((size_t)b * CIN + (size_t)a * OC) * HW + hw;
  float* dst = xbuf + (size_t)p * OC;
#pragma unroll
  for (int c = 0; c < OC; ++c) dst[c] = src[(size_t)c * HW];
}

// ---------------------------------------------------------------------------
// Conv3x3 32->32 as implicit GEMM with V_WMMA_F32_16X16X4_F32.
// in/out are NHWC(32).  affine (scale[32],shift[32]) applied to in-bounds
// inputs (identity when use_affine==0).  Per-block BN partial sums written
// deterministically to partial[block*64 + {sum[32],sq[32]}].
// ---------------------------------------------------------------------------
__global__ __launch_bounds__(128) void conv_k(const float* __restrict__ in,
                                              const float* __restrict__ wgt,
                                              const float* __restrict__ bias,
                                              const float* __restrict__ affine,
                                              int use_affine,
                                              float* __restrict__ out,
                                              float* __restrict__ partial) {
  __shared__ float lds_w[288 * 32];              // [k=tap*32+ci][o]
  __shared__ float lds_x[3][32][XSTRIDE];        // [row][ci][col]  col = x - (x0-1)
  __shared__ float lds_aff[64];

  const int tid  = threadIdx.x;
  const int blk  = blockIdx.x;                   // 0..1023
  const int tX   = blk & 1;
  const int y    = (blk >> 1) & (HH - 1);
  const int b    = blk >> 8;
  const int x0   = tX * TILEW;

  if (tid < 64)
    lds_aff[tid] = use_affine ? affine[tid] : (tid < 32 ? 1.0f : 0.0f);
  __syncthreads();

  // stage weights: w[o][ci][ty][tx] -> lds_w[(tap*32+ci)*32 + o]
  for (int i = tid; i < 9216; i += 128) {
    int tap = i % 9;
    int oci = i / 9;
    int ci  = oci & 31;
    int o   = oci >> 5;
    lds_w[(tap * 32 + ci) * 32 + o] = wgt[i];
  }
  // stage 3 input rows x [x0-1, x0+64], affine applied, zero pad AFTER affine
  for (int i = tid; i < 3 * 66 * 32; i += 128) {
    int c   = i & 31;
    int col = (i >> 5) % 66;
    int r   = i / (66 * 32);
    int yy  = y + r - 1;
    int xx  = x0 + col - 1;
    float v = 0.0f;
    if ((unsigned)yy < (unsigned)HH && (unsigned)xx < (unsigned)WW) {
      v = in[(((size_t)(b * HH + yy) * WW + xx) * OC) + c];
      v = v * lds_aff[c] + lds_aff[32 + c];
    }
    lds_x[r][c][col] = v;
  }
  __syncthreads();

  const int wave = tid >> 5;
  const int lane = tid & 31;
  const int pos  = lane & 15;
  const int half = lane >> 4;
  const int wx0  = wave * 16;

  v8f acc0 = {};
  v8f acc1 = {};
  for (int tap = 0; tap < 9; ++tap) {
    const int ty = tap / 3;
    const int tx = tap % 3;
    const int colbase = wx0 + pos + tx;          // <= 65 < XSTRIDE
#pragma unroll
    for (int cb = 0; cb < 32; cb += 4) {
      const int c0i = cb + 2 * half;
      v2f af;
      af.x = lds_x[ty][c0i][colbase];
      af.y = lds_x[ty][c0i + 1][colbase];
      const int kidx = (tap * 32 + c0i) * 32;
      v2f bf0, bf1;
      bf0.x = lds_w[kidx + pos];
      bf0.y = lds_w[kidx + 32 + pos];
      bf1.x = lds_w[kidx + 16 + pos];
      bf1.y = lds_w[kidx + 48 + pos];
      acc0 = __builtin_amdgcn_wmma_f32_16x16x4_f32(false, af, false, bf0,
                                                   (short)0, acc0, false, false);
      acc1 = __builtin_amdgcn_wmma_f32_16x16x4_f32(false, af, false, bf1,
                                                   (short)0, acc1, false, false);
    }
  }

  // epilogue: +bias, store NHWC, local BN sums
  const int n0 = pos, n1 = pos + 16;
  const float bv0 = bias[n0], bv1 = bias[n1];
  float s0 = 0.f, q0 = 0.f, s1 = 0.f, q1 = 0.f;
  const size_t pbase = (size_t)(b * HH + y) * WW + (x0 + wx0 + 8 * half);
#pragma unroll
  for (int v = 0; v < 8; ++v) {
    float y0 = acc0[v] + bv0;
    float y1 = acc1[v] + bv1;
    out[(pbase + v) * OC + n0] = y0;
    out[(pbase + v) * OC + n1] = y1;
    s0 += y0; q0 += y0 * y0;
    s1 += y1; q1 += y1 * y1;
  }

  // deterministic block reduction: reuse lds_x as scratch (done reading it)
  __syncthreads();
  float* red = &lds_x[0][0][0];                  // 8 groups x 64 floats
  const int g = wave * 2 + half;
  red[g * 64 + pos]      = s0;
  red[g * 64 + 16 + pos] = s1;
  red[g * 64 + 32 + pos] = q0;
  red[g * 64 + 48 + pos] = q1;
  __syncthreads();
  if (tid < 64) {
    float acc = 0.f;
#pragma unroll
    for (int gg = 0; gg < 8; ++gg) acc += red[gg * 64 + tid];
    partial[(size_t)blk * 64 + tid] = acc;
  }
}

// ---------------------------------------------------------------------------
// Reduce 1024 block partials -> per-channel BN affine (scale, shift)
// ---------------------------------------------------------------------------
__global__ __launch_bounds__(32) void finalize_k(const float* __restrict__ partial,
                                                 const float* __restrict__ gamma,
                                                 const float* __restrict__ beta,
                                                 float* __restrict__ affine) {
  int c = threadIdx.x;                           // 0..31
  float s = 0.f, q = 0.f;
  for (int blk = 0; blk < NBLK_CONV; ++blk) {
    s += partial[(size_t)blk * 64 + c];
    q += partial[(size_t)blk * 64 + 32 + c];
  }
  const float inv_n = 1.0f / (float)NPIX;
  float mean = s * inv_n;
  float var  = q * inv_n - mean * mean;
  float sc   = gamma[c] * rsqrtf(var + 1e-5f);
  affine[c]      = sc;
  affine[32 + c] = beta[c] - mean * sc;
}

// ---------------------------------------------------------------------------
// Apply BN2 affine and scatter back into the orientation-selected channels
// ---------------------------------------------------------------------------
__global__ __launch_bounds__(256) void scatter_k(const float* __restrict__ ybuf,
                                                 const int* __restrict__ idxb,
                                                 const float* __restrict__ affine,
                                                 float* __restrict__ outp) {
  __shared__ float aff[64];
  if (threadIdx.x < 64) aff[threadIdx.x] = affine[threadIdx.x];
  __syncthreads();
  int p  = blockIdx.x * 256 + threadIdx.x;
  int b  = p >> 14;
  int hw = p & 16383;
  int a  = idxb[p];
  float* dst = outp + ((size_t)b * CIN + (size_t)a * OC) * HW + hw;
  const float* src = ybuf + (size_t)p * OC;
#pragma unroll
  for (int c = 0; c < OC; ++c)
    dst[(size_t)c * HW] = src[c] * aff[c] + aff[32 + c];
}

// ---------------------------------------------------------------------------
extern "C" void kernel_launch(void* const* d_in, const int* in_sizes, int n_in,
                              void* d_out, int out_size, void* d_ws, size_t ws_size,
                              hipStream_t stream) {
  const float* feats = (const float*)d_in[0];
  const float* rois  = (const float*)d_in[1];
  const float* w1    = (const float*)d_in[2];
  const float* b1    = (const float*)d_in[3];
  const float* g1    = (const float*)d_in[4];
  const float* be1   = (const float*)d_in[5];
  const float* w2    = (const float*)d_in[6];
  const float* b2    = (const float*)d_in[7];
  const float* g2    = (const float*)d_in[8];
  const float* be2   = (const float*)d_in[9];
  float* out = (float*)d_out;

  char* ws = (char*)d_ws;
  const size_t BUF = (size_t)NPIX * OC * sizeof(float);      // 8 MiB
  float* buf0    = (float*)(ws);                              // gathered x, then y2
  float* buf1    = (float*)(ws + BUF);                        // y1
  int*   idxb    = (int*)  (ws + 2 * BUF);                    // 256 KiB
  float* partial = (float*)(ws + 2 * BUF + (size_t)NPIX * 4); // 1024*64 floats
  float* aff1    = (float*)(ws + 2 * BUF + (size_t)NPIX * 4 + (size_t)NBLK_CONV * 64 * 4);
  float* aff2    = aff1 + 64;

  // untouched channels: straight copy of feats into the output
  hipMemcpyAsync(out, feats, (size_t)BATCH * CIN * HW * sizeof(float),
                 hipMemcpyDeviceToDevice, stream);

  gather_k<<<NPIX / 256, 256, 0, stream>>>(feats, rois, buf0, idxb);

  conv_k<<<NBLK_CONV, 128, 0, stream>>>(buf0, w1, b1, aff1, 0, buf1, partial);
  finalize_k<<<1, 32, 0, stream>>>(partial, g1, be1, aff1);

  conv_k<<<NBLK_CONV, 128, 0, stream>>>(buf1, w2, b2, aff1, 1, buf0, partial);
  finalize_k<<<1, 32, 0, stream>>>(partial, g2, be2, aff2);

  scatter_k<<<NPIX / 256, 256, 0, stream>>>(buf0, idxb, aff2, out);
}